// PositionCollapseLoss_16801912062785
// MI455X (gfx1250) — compile-verified
//
#include <hip/hip_runtime.h>
#include <hip/hip_bf16.h>
#include <math.h>

typedef __attribute__((ext_vector_type(2))) float v2f;
typedef __attribute__((ext_vector_type(8))) float v8f;

#define NBATCH  16
#define NCH     14
#define THREADS 256
#define WAVES   8   // 256 threads / wave32

// ---------------------------------------------------------------------------
// Kernel 1: per-block partial Gram matrix + channel sums via V_WMMA_F32_16X16X4_F32
//
// D = A(16x4)*B(4x16)+C per 4-point K-slice:
//   A(m,k) = channel m of point k (m<3), A(3,k)=1 (ones row), rows>3 don't care
//   B(k,n) = channel n of point k (n<3), B(k,3)=1,            cols>3 don't care
// A and B have identical per-lane register images here, so one v2f feeds both.
// Accumulated D(0..2,0..2) = Gram block, D(0..2,3) = channel sums.
// Branchless lane value: v = load * chMask + fillAdd  (fma, no exec games).
// ---------------------------------------------------------------------------
__global__ __launch_bounds__(THREADS) void cov_partial_kernel(
    const float* __restrict__ g, float* __restrict__ partials, int N, int bpb)
{
  const int batch = blockIdx.x / bpb;
  const int blk   = blockIdx.x % bpb;
  const float* __restrict__ gb = g + (size_t)batch * (size_t)N * NCH;

  const int tid  = threadIdx.x;
  const int w    = tid >> 5;
  const int lane = tid & 31;
  const int m    = lane & 15;   // matrix row (A) / col (B) index for this lane
  const int half = lane >> 4;   // selects K pair {0,1} vs {2,3}

  const int ppb   = (N + bpb - 1) / bpb;        // points per block
  const int ppw   = (ppb + WAVES - 1) / WAVES;  // points per wave
  const int start = blk * ppb + w * ppw;
  int limit = blk * ppb + ppb;
  if (limit > N) limit = N;

  const int   ch      = (m < 3) ? m : 0;          // clamped channel -> safe addr
  const float chMask  = (m < 3) ? 1.0f : 0.0f;
  const float fillAdd = (m == 3) ? 1.0f : 0.0f;   // ones row/col augmentation

  // This wave's in-bounds point count, split into unchecked main + guarded tail.
  int n = limit - start;
  if (n < 0)   n = 0;
  if (n > ppw) n = ppw;
  const int nmain = n & ~15;                      // 16 points (4 WMMAs) per trip

  v8f acc0 = {};
  v8f acc1 = {};

  // Hot loop: no bounds checks, no branches besides the loop back-edge.
  // Lane-private base: this lane's K=0 point of group 0.
  const float* __restrict__ lp = gb + (size_t)(start + half * 2) * NCH + ch;
  for (int it = 0; it < nmain; it += 16) {
#pragma unroll
    for (int u = 0; u < 4; ++u) {
      const int o = (it + u * 4) * NCH;
      const float l0 = __builtin_nontemporal_load(lp + o);        // K even
      const float l1 = __builtin_nontemporal_load(lp + o + NCH);  // K odd
      v2f ab;
      ab.x = fmaf(l0, chMask, fillAdd);
      ab.y = fmaf(l1, chMask, fillAdd);
      if (u & 1)
        acc1 = __builtin_amdgcn_wmma_f32_16x16x4_f32(false, ab, false, ab,
                                                     (short)0, acc1, false, false);
      else
        acc0 = __builtin_amdgcn_wmma_f32_16x16x4_f32(false, ab, false, ab,
                                                     (short)0, acc0, false, false);
    }
  }

  // Guarded tail (never runs for the reference shape: 1024/8 = 128 = 8*16).
  for (int it = nmain; it < n; it += 4) {
    const int p  = start + it + half * 2;
    const int q  = p + 1;
    const int pc = (p < limit) ? p : (limit - 1);
    const int qc = (q < limit) ? q : (limit - 1);
    const float l0 = gb[pc * NCH + ch];
    const float l1 = gb[qc * NCH + ch];
    const float m0 = (p < limit) ? chMask : 0.0f;
    const float m1 = (q < limit) ? chMask : 0.0f;
    v2f ab;
    ab.x = fmaf(l0, m0, fillAdd);
    ab.y = fmaf(l1, m1, fillAdd);
    acc0 = __builtin_amdgcn_wmma_f32_16x16x4_f32(false, ab, false, ab,
                                                 (short)0, acc0, false, false);
  }

  // D layout: element r = VGPR r = row M=r (lanes 0-15), col N=lane.
  // Need rows 0..2, cols 0..3 -> lanes 0..3, elements 0..2.
  __shared__ float red[WAVES][12];
  if (lane < 4) {
    red[w][lane * 3 + 0] = acc0[0] + acc1[0];
    red[w][lane * 3 + 1] = acc0[1] + acc1[1];
    red[w][lane * 3 + 2] = acc0[2] + acc1[2];
  }
  __syncthreads();
  if (tid < 12) {
    float s = 0.0f;
#pragma unroll
    for (int ww = 0; ww < WAVES; ++ww) s += red[ww][tid];
    // j = col*3 + row: 0:Sxx 1:Sxy 2:Sxz 3:Sxy 4:Syy 5:Syz 6:Sxz 7:Syz 8:Szz 9:Sx 10:Sy 11:Sz
    partials[(size_t)blockIdx.x * 12 + tid] = s;
  }
}

// ---------------------------------------------------------------------------
// Kernel 2: deterministic f64 reduction of partials + closed-form symmetric
// 3x3 eigenvalues (Smith's trigonometric method) + final mean.
// ---------------------------------------------------------------------------
__global__ __launch_bounds__(256) void finalize_kernel(
    const float* __restrict__ partials, float* __restrict__ out, int N, int bpb)
{
  __shared__ double sums[NBATCH][12];
  const int t = threadIdx.x;
  if (t < NBATCH * 12) {
    const int b = t / 12, j = t % 12;
    const float* p = partials + (size_t)b * bpb * 12 + j;
    double s = 0.0;
    for (int i = 0; i < bpb; ++i) s += (double)p[(size_t)i * 12];
    sums[b][j] = s;
  }
  __syncthreads();

  if (t == 0) {
    const double invN = 1.0 / (double)N;
    double accv = 0.0;
    for (int b = 0; b < NBATCH; ++b) {
      const double* S = sums[b];
      const double mx = S[9] * invN, my = S[10] * invN, mz = S[11] * invN;
      const double a00 = S[0] * invN - mx * mx;
      const double a01 = 0.5 * (S[1] + S[3]) * invN - mx * my;
      const double a02 = 0.5 * (S[2] + S[6]) * invN - mx * mz;
      const double a11 = S[4] * invN - my * my;
      const double a12 = 0.5 * (S[5] + S[7]) * invN - my * mz;
      const double a22 = S[8] * invN - mz * mz;

      const double p1 = a01 * a01 + a02 * a02 + a12 * a12;
      const double qd = (a00 + a11 + a22) / 3.0;
      const double b00 = a00 - qd, b11 = a11 - qd, b22 = a22 - qd;
      const double p2 = b00 * b00 + b11 * b11 + b22 * b22 + 2.0 * p1;
      const double pr = sqrt(p2 / 6.0);

      double emax, emin;
      if (pr < 1e-30) {
        emax = qd; emin = qd;
      } else {
        const double ip  = 1.0 / pr;
        const double c00 = b00 * ip, c01 = a01 * ip, c02 = a02 * ip;
        const double c11 = b11 * ip, c12 = a12 * ip, c22 = b22 * ip;
        double detB = c00 * (c11 * c22 - c12 * c12)
                    - c01 * (c01 * c22 - c12 * c02)
                    + c02 * (c01 * c12 - c11 * c02);
        double r = detB * 0.5;
        r = fmin(1.0, fmax(-1.0, r));
        const double phi = acos(r) / 3.0;
        emax = qd + 2.0 * pr * cos(phi);
        emin = qd + 2.0 * pr * cos(phi + 2.0943951023931953);  // + 2*pi/3
      }
      accv += -(emax / (emin + 1e-6));
    }
    out[0] = (float)(accv / (double)NBATCH);
  }
}

// ---------------------------------------------------------------------------
extern "C" void kernel_launch(void* const* d_in, const int* in_sizes, int n_in,
                              void* d_out, int out_size, void* d_ws, size_t ws_size,
                              hipStream_t stream)
{
  const float* g   = (const float*)d_in[0];
  float* out       = (float*)d_out;
  float* partials  = (float*)d_ws;

  const int N = in_sizes[0] / (NBATCH * NCH);   // 262144 for the reference

  // Blocks per batch: as many as workspace allows, capped at 256.
  int bpb = 256;
  while (bpb > 1 && (size_t)NBATCH * (size_t)bpb * 12 * sizeof(float) > ws_size)
    bpb >>= 1;

  cov_partial_kernel<<<dim3(NBATCH * bpb), THREADS, 0, stream>>>(g, partials, N, bpb);
  finalize_kernel<<<dim3(1), 256, 0, stream>>>(partials, out, N, bpb);
}